// TopologicalSignatureDistance_71382356460207
// MI455X (gfx1250) — compile-verified
//
#include <hip/hip_runtime.h>
#include <hip/hip_bf16.h>
#include <float.h>

#define NN 4096
#define NT 1024

typedef float v2f __attribute__((ext_vector_type(2)));
typedef float v8f __attribute__((ext_vector_type(8)));

// ---------------------------------------------------------------------------
// Kernel 1: Prim's MST, one workgroup per matrix (blockIdx.x selects matrix).
// min_dist/min_src live in registers: thread t owns nodes t, t+1024, t+2048,
// t+3072 (coalesced row reads). Per iteration: masked register argmin ->
// wave32 shuffle reduction -> 32-partial LDS reduction by wave 0 -> broadcast
// v -> owner thread publishes min_src[v] -> all threads update from row W[v]
// (L2-resident: both 64MB matrices fit in the 192MB L2).
// Emits per-edge weight (f32) and packed pair key u*4096+v (u<v) to scratch.
// ---------------------------------------------------------------------------
__global__ __launch_bounds__(NT) void prim_kernel(const float* __restrict__ D0,
                                                  const float* __restrict__ D1,
                                                  float* __restrict__ w_out,
                                                  unsigned* __restrict__ k_out) {
  const float* __restrict__ D = (blockIdx.x == 0) ? D0 : D1;
  float*    ew = w_out + (size_t)blockIdx.x * NN;
  unsigned* ek = k_out + (size_t)blockIdx.x * NN;

  const int t = threadIdx.x;
  float md[4];
  int   ms[4];
  bool  it[4];
#pragma unroll
  for (int k = 0; k < 4; ++k) {
    int node = t + k * NT;
    md[k] = D[node];      // row 0 of W (W == D: symmetric, zero diagonal)
    ms[k] = 0;
    it[k] = (node == 0);  // node 0 seeds the tree
  }

  __shared__ float s_val[32];
  __shared__ int   s_idx[32];
  __shared__ int   s_v;
  __shared__ int   s_u;
  __shared__ float s_w;

  const int lane = t & 31;
  const int wave = t >> 5;

  for (int i = 0; i < NN - 1; ++i) {
    // --- local masked argmin over the 4 owned nodes ---
    float bv = FLT_MAX;
    int   bi = 0;
#pragma unroll
    for (int k = 0; k < 4; ++k) {
      if (!it[k] && md[k] < bv) { bv = md[k]; bi = t + k * NT; }
    }
    // --- wave32 min-reduction ---
#pragma unroll
    for (int off = 16; off >= 1; off >>= 1) {
      float ov = __shfl_down(bv, off, 32);
      int   oi = __shfl_down(bi, off, 32);
      if (ov < bv) { bv = ov; bi = oi; }
    }
    if (lane == 0) { s_val[wave] = bv; s_idx[wave] = bi; }
    __syncthreads();
    // --- wave 0 reduces the 32 per-wave partials ---
    if (wave == 0) {
      float v2 = s_val[lane];
      int   i2 = s_idx[lane];
#pragma unroll
      for (int off = 16; off >= 1; off >>= 1) {
        float ov = __shfl_down(v2, off, 32);
        int   oi = __shfl_down(i2, off, 32);
        if (ov < v2) { v2 = ov; i2 = oi; }
      }
      if (lane == 0) { s_v = i2; s_w = v2; }
    }
    __syncthreads();
    const int v = s_v;
    // owner of v publishes min_src[v] and marks v in-tree
    if ((v & (NT - 1)) == t) {
      int kk = v >> 10;
      s_u = ms[kk];
      it[kk] = true;
    }
    __syncthreads();
    if (t == 0) {
      int u = s_u;
      int a = u < v ? u : v;
      int b = u < v ? v : u;
      ew[i] = s_w;                               // == W[a][b]
      ek[i] = (unsigned)a * NN + (unsigned)b;    // fits in 24 bits
    }
    // --- frontier update from row W[v] (coalesced, L2-resident) ---
    const float* __restrict__ row = D + (size_t)v * NN;
#pragma unroll
    for (int k = 0; k < 4; ++k) {
      float wv = row[t + k * NT];
      if (wv < md[k]) { md[k] = wv; ms[k] = v; }
    }
    __syncthreads();
  }
  // pad slot 4095: weight 0.0 (sorts to front in BOTH arrays -> diff 0),
  // key 0xFFFFFFFF (sorts past all real 24-bit keys -> excluded from match)
  if (t == 0) { ew[NN - 1] = 0.0f; ek[NN - 1] = 0xFFFFFFFFu; }
}

// ---------------------------------------------------------------------------
// Kernel 2: in-LDS bitonic sort of one 4096-element u32 array per block.
// Weights are non-negative floats, so their IEEE bit patterns sort correctly
// as u32. grid = 4: {weights m0, weights m1, keys m0, keys m1}.
// ---------------------------------------------------------------------------
__global__ __launch_bounds__(NT) void sort_kernel(unsigned* __restrict__ data) {
  __shared__ unsigned s[NN];
  unsigned* arr = data + (size_t)blockIdx.x * NN;
  const int t = threadIdx.x;
#pragma unroll
  for (int m = 0; m < 4; ++m) s[t + m * NT] = arr[t + m * NT];
  __syncthreads();
  for (unsigned k = 2; k <= NN; k <<= 1) {
    for (unsigned j = k >> 1; j > 0; j >>= 1) {
#pragma unroll
      for (int m = 0; m < 4; ++m) {
        unsigned i = (unsigned)t + (unsigned)m * NT;
        unsigned ixj = i ^ j;
        if (ixj > i) {
          unsigned a = s[i], b = s[ixj];
          bool up = ((i & k) == 0);
          if (up ? (a > b) : (a < b)) { s[i] = b; s[ixj] = a; }
        }
      }
      __syncthreads();
    }
  }
#pragma unroll
  for (int m = 0; m < 4; ++m) arr[t + m * NT] = s[t + m * NT];
}

// ---------------------------------------------------------------------------
// Kernel 3: finalize.
//  - matched: each of 4095 keys from set 1 binary-searched in sorted set 2.
//  - distance: sum_i (sig1[i]-sig2[i])^2 accumulated on the matrix engine:
//    V_WMMA_F32_16X16X4_F32 with B = all-ones => D[m][n] += sum_k A[m][k].
//    One full wave (EXEC all-ones) consumes 64 squared diffs per WMMA,
//    64 WMMA issues total; final 16 column sums folded via shuffles.
// ---------------------------------------------------------------------------
__global__ __launch_bounds__(256) void finalize_kernel(const float* __restrict__ sig1,
                                                       const float* __restrict__ sig2,
                                                       const unsigned* __restrict__ k1,
                                                       const unsigned* __restrict__ k2,
                                                       float* __restrict__ out) {
  __shared__ int s_cnt;
  if (threadIdx.x == 0) s_cnt = 0;
  __syncthreads();

  int local = 0;
  for (int i = threadIdx.x; i < NN - 1; i += blockDim.x) {
    unsigned key = k1[i];
    int lo = 0, hi = NN - 2;  // real keys occupy [0, 4095); pad is at 4095
    while (lo <= hi) {
      int mid = (lo + hi) >> 1;
      unsigned vv = k2[mid];
      if (vv == key) { ++local; break; }
      if (vv < key) lo = mid + 1; else hi = mid - 1;
    }
  }
  atomicAdd(&s_cnt, local);
  __syncthreads();
  if (threadIdx.x == 0) out[1] = (float)s_cnt;

  // ---- WMMA reduction by wave 0 (fully active => EXEC == 0xFFFFFFFF) ----
  if (threadIdx.x < 32) {
    const int lane = threadIdx.x;
    v8f c = {};            // f32 16x16 accumulator, 8 VGPRs/lane
    v2f ones;
    ones.x = 1.0f; ones.y = 1.0f;
    for (int itn = 0; itn < NN / 64; ++itn) {
      int base = itn * 64 + lane * 2;
      float d0 = sig1[base] - sig2[base];
      float d1 = sig1[base + 1] - sig2[base + 1];
      v2f a;
      a.x = d0 * d0;
      a.y = d1 * d1;
      // D = A(16x4) * ones(4x16) + C : every column of D holds the row sums
      c = __builtin_amdgcn_wmma_f32_16x16x4_f32(
          /*neg_a=*/false, a, /*neg_b=*/false, ones,
          /*c_mod=*/(short)0, c, /*reuse_a=*/false, /*reuse_b=*/false);
    }
    float s = c[0] + c[1] + c[2] + c[3] + c[4] + c[5] + c[6] + c[7];
    // column 0: M=0..7 -> lane 0, M=8..15 -> lane 16
    float tot = __shfl(s, 0, 32) + __shfl(s, 16, 32);
    if (lane == 0) out[0] = tot;
  }
}

// ---------------------------------------------------------------------------
extern "C" void kernel_launch(void* const* d_in, const int* in_sizes, int n_in,
                              void* d_out, int out_size, void* d_ws, size_t ws_size,
                              hipStream_t stream) {
  const float* D0 = (const float*)d_in[0];
  const float* D1 = (const float*)d_in[1];
  float* out = (float*)d_out;

  // workspace layout (u32 units): [0,4096) w_m0 | [4096,8192) w_m1 |
  //                               [8192,12288) key_m0 | [12288,16384) key_m1
  float*    w_buf = (float*)d_ws;
  unsigned* k_buf = (unsigned*)d_ws + 2 * NN;

  prim_kernel<<<2, NT, 0, stream>>>(D0, D1, w_buf, k_buf);
  sort_kernel<<<4, NT, 0, stream>>>((unsigned*)d_ws);
  finalize_kernel<<<1, 256, 0, stream>>>(w_buf, w_buf + NN, k_buf, k_buf + NN, out);
}